// KGMC_SAGE_77919296684697
// MI455X (gfx1250) — compile-verified
//
#include <hip/hip_runtime.h>
#include <hip/hip_bf16.h>

typedef __attribute__((ext_vector_type(2))) float v2f;
typedef __attribute__((ext_vector_type(8))) float v8f;

#define NN 200000
#define NE 3200000
#define NP 4096

// D = A(16x4 f32) x B(4x16 f32) + C(16x16 f32), wave32.
static __device__ __forceinline__ v8f wmma4(v2f a, v2f b, v8f c) {
  return __builtin_amdgcn_wmma_f32_16x16x4_f32(
      /*neg_a=*/false, a, /*neg_b=*/false, b,
      /*c_mod=*/(short)0, c, /*reuse_a=*/false, /*reuse_b=*/false);
}

__global__ void zero_f32_kernel(float* __restrict__ p, size_t n) {
  size_t i = (size_t)blockIdx.x * blockDim.x + threadIdx.x;
  size_t stride = (size_t)gridDim.x * blockDim.x;
  for (; i < n; i += stride) p[i] = 0.0f;
}

__global__ void degree_kernel(const int* __restrict__ dst, float* __restrict__ deg, int ne) {
  int e = blockIdx.x * blockDim.x + threadIdx.x;
  if (e < ne) atomicAdd(&deg[dst[e]], 1.0f);
}

// One lane per (edge, feature): coalesced 32-float row gather + float atomic scatter.
__global__ void scatter_kernel(const float* __restrict__ hin, int hstride,
                               const int* __restrict__ src, const int* __restrict__ dst,
                               float* __restrict__ agg, int ne) {
  long long tid = (long long)blockIdx.x * blockDim.x + threadIdx.x;
  int e = (int)(tid >> 5);
  if (e >= ne) return;
  int f = (int)(tid & 31);
  int s = src[e];
  int d = dst[e];
  float v = hin[(size_t)s * hstride + f];
  atomicAdd(&agg[(size_t)d * 32 + f], v);
}

// One wave per 16-node tile:
//   Hout[16,32] = tanh(Hin[16,32] @ Ws + Mean[16,32] @ Wn + b)
// via chained v_wmma_f32_16x16x4_f32 (8 K-steps x 2 operands x 2 col-blocks).
__global__ void sage_update_kernel(const float* __restrict__ hin, int hstride,
                                   const float* __restrict__ agg,
                                   const float* __restrict__ deg,
                                   const float* __restrict__ Ws,
                                   const float* __restrict__ Wn,
                                   const float* __restrict__ bias,
                                   float* __restrict__ states, int layer, int ntiles) {
  int wave = blockIdx.x * (blockDim.x >> 5) + ((int)threadIdx.x >> 5);
  if (wave >= ntiles) return;                 // wave-uniform: WMMA below runs with EXEC all-ones
  int lane = (int)threadIdx.x & 31;
  int lm = lane & 15;                         // A row (M) held by this lane
  int half = lane >> 4;                       // K sub-pair select (0: K=0,1  1: K=2,3)
  int node0 = wave << 4;

  const float* hrow = hin + (size_t)(node0 + lm) * hstride;
  const float* arow = agg + (size_t)(node0 + lm) * 32;
  float inv = 1.0f / fmaxf(deg[node0 + lm], 1.0f);

  for (int nb = 0; nb < 2; ++nb) {
    int col = nb * 16 + lm;                   // B/C/D column (N) held by this lane
    float bv = bias[col];
    v8f acc;
#pragma unroll
    for (int i = 0; i < 8; ++i) acc[i] = bv;
#pragma unroll
    for (int kb = 0; kb < 8; ++kb) {
      int k0 = kb * 4 + half * 2;
      v2f ah; ah[0] = hrow[k0];            ah[1] = hrow[k0 + 1];
      v2f am; am[0] = arow[k0] * inv;      am[1] = arow[k0 + 1] * inv;
      v2f bs; bs[0] = Ws[(size_t)k0 * 32 + col]; bs[1] = Ws[(size_t)(k0 + 1) * 32 + col];
      v2f bn; bn[0] = Wn[(size_t)k0 * 32 + col]; bn[1] = Wn[(size_t)(k0 + 1) * 32 + col];
      acc = wmma4(ah, bs, acc);
      acc = wmma4(am, bn, acc);
    }
#pragma unroll
    for (int i = 0; i < 8; ++i) {
      int row = i + half * 8;                 // C/D: lanes 0-15 -> M=i, lanes 16-31 -> M=i+8
      states[(size_t)(node0 + row) * 128 + layer * 32 + col] = tanhf(acc[i]);
    }
  }
}

// One wave per 16 pairs: hdn[16,128] = relu([cs[u] | cs[i]] @ W1 + bl1)
__global__ void pair_gemm_kernel(const float* __restrict__ states,
                                 const int* __restrict__ uidx, const int* __restrict__ iidx,
                                 const float* __restrict__ W1, const float* __restrict__ bl1,
                                 float* __restrict__ hdn, int ntiles) {
  int wave = blockIdx.x * (blockDim.x >> 5) + ((int)threadIdx.x >> 5);
  if (wave >= ntiles) return;
  int lane = (int)threadIdx.x & 31;
  int lm = lane & 15;
  int half = lane >> 4;
  int p0 = wave << 4;

  const float* urow = states + (size_t)uidx[p0 + lm] * 128;
  const float* irow = states + (size_t)iidx[p0 + lm] * 128;

  for (int nb = 0; nb < 8; ++nb) {
    int col = nb * 16 + lm;
    float bv = bl1[col];
    v8f acc;
#pragma unroll
    for (int i = 0; i < 8; ++i) acc[i] = bv;
    for (int kb = 0; kb < 64; ++kb) {
      int k0 = kb * 4 + half * 2;             // k0 even; k0,k0+1 always in same half of 256
      const float* r = (k0 < 128) ? urow : irow;
      int kk = k0 & 127;
      v2f a; a[0] = r[kk]; a[1] = r[kk + 1];
      v2f b; b[0] = W1[(size_t)k0 * 128 + col]; b[1] = W1[(size_t)(k0 + 1) * 128 + col];
      acc = wmma4(a, b, acc);
    }
#pragma unroll
    for (int i = 0; i < 8; ++i) {
      int row = i + half * 8;
      hdn[(size_t)(p0 + row) * 128 + col] = fmaxf(acc[i], 0.0f);
    }
  }
}

__global__ void pair_out_kernel(const float* __restrict__ hdn, const float* __restrict__ W2,
                                const float* __restrict__ bl2, float* __restrict__ score, int np) {
  int p = blockIdx.x * blockDim.x + threadIdx.x;
  if (p >= np) return;
  float s = bl2[0];
  const float* h = hdn + (size_t)p * 128;
#pragma unroll 8
  for (int k = 0; k < 128; ++k) s += h[k] * W2[k];
  score[p] = 1.0f / (1.0f + expf(-s));        // MULTIPLY_BY == 1.0
}

extern "C" void kernel_launch(void* const* d_in, const int* in_sizes, int n_in,
                              void* d_out, int out_size, void* d_ws, size_t ws_size,
                              hipStream_t stream) {
  (void)in_sizes; (void)n_in; (void)out_size; (void)ws_size;

  const float* x    = (const float*)d_in[0];
  const int*   src  = (const int*)d_in[1];
  const int*   dst  = (const int*)d_in[2];
  const int*   uidx = (const int*)d_in[3];
  const int*   iidx = (const int*)d_in[4];
  const float* Ws[4] = {(const float*)d_in[5],  (const float*)d_in[8],
                        (const float*)d_in[11], (const float*)d_in[14]};
  const float* Wn[4] = {(const float*)d_in[6],  (const float*)d_in[9],
                        (const float*)d_in[12], (const float*)d_in[15]};
  const float* bb[4] = {(const float*)d_in[7],  (const float*)d_in[10],
                        (const float*)d_in[13], (const float*)d_in[16]};
  const float* W1  = (const float*)d_in[17];
  const float* bl1 = (const float*)d_in[18];
  const float* W2  = (const float*)d_in[19];
  const float* bl2 = (const float*)d_in[20];

  float* out    = (float*)d_out;
  float* score  = out;                 // [4096]
  float* states = out + NP;            // [200000, 128] concat of 4 layer outputs

  float* ws  = (float*)d_ws;
  float* deg = ws;                                   // [NN]
  float* agg = ws + NN;                              // [NN, 32]
  float* hdn = agg + (size_t)NN * 32;                // [NP, 128]

  // degree (shared across layers) + first agg zeroing
  zero_f32_kernel<<<2048, 256, 0, stream>>>(deg, (size_t)NN + (size_t)NN * 32);
  degree_kernel<<<(NE + 255) / 256, 256, 0, stream>>>(dst, deg, NE);

  const int ntiles = NN / 16;                        // 12500
  const int ublocks = (ntiles + 7) / 8;              // 8 waves / 256-thread block
  const long long sthreads = (long long)NE * 32;     // 102.4M lanes
  const int sblocks = (int)((sthreads + 255) / 256);

  for (int l = 0; l < 4; ++l) {
    if (l > 0)
      zero_f32_kernel<<<2048, 256, 0, stream>>>(agg, (size_t)NN * 32);
    const float* hin = (l == 0) ? x : (states + (size_t)(l - 1) * 32);
    int hstride = (l == 0) ? 32 : 128;
    scatter_kernel<<<sblocks, 256, 0, stream>>>(hin, hstride, src, dst, agg, NE);
    sage_update_kernel<<<ublocks, 256, 0, stream>>>(hin, hstride, agg, deg,
                                                    Ws[l], Wn[l], bb[l],
                                                    states, l, ntiles);
  }

  pair_gemm_kernel<<<(NP / 16 + 7) / 8, 256, 0, stream>>>(states, uidx, iidx, W1, bl1,
                                                          hdn, NP / 16);
  pair_out_kernel<<<(NP + 255) / 256, 256, 0, stream>>>(hdn, W2, bl2, score, NP);
}